// FrameWiseCrossAttention_55628416418485
// MI455X (gfx1250) — compile-verified
//
#include <hip/hip_runtime.h>

typedef float v2f __attribute__((ext_vector_type(2)));
typedef float v4f __attribute__((ext_vector_type(4)));
typedef float v8f __attribute__((ext_vector_type(8)));

#define DIMSZ   1024
#define BT      32          // B * T rows in the small GEMMs
#define TFRAMES 16
#define TPF     1560
#define NPB     (TFRAMES * TPF)   // 24960 tokens per batch

// ---------------------------------------------------------------------------
// One wave computes one 16x16 tile of  OUT = A(32xK) * W^T + bias, K = 1024,
// using V_WMMA_F32_16X16X4_F32 (full fp32 precision, fp32 accumulate).
// Grid: 16 blocks x 256 threads = 128 waves = 128 tiles (2 in M, 64 in N).
// ---------------------------------------------------------------------------
__global__ __launch_bounds__(256) void gemm32_wmma_f32(
    const float* __restrict__ A,     // [32, 1024] row-major
    const float* __restrict__ W,     // [1024, 1024] row-major; used as B = W^T
    const float* __restrict__ bias,  // [1024]
    float* __restrict__ out)         // [32, 1024] row-major
{
    const int lane = threadIdx.x & 31;
    const int wave = threadIdx.x >> 5;
    const int tile = blockIdx.x * 8 + wave;   // 0..127
    const int m0 = (tile & 1) * 16;           // 2 tiles along M (32 rows)
    const int n0 = (tile >> 1) * 16;          // 64 tiles along N (1024 cols)

    const int half = lane >> 4;               // 0: K pair {k,k+1}, 1: {k+2,k+3}
    const int l15  = lane & 15;

    // A fragment: lane holds row m0+l15, contiguous float2 at k + 2*half.
    // B fragment (B = W^T): identical pattern on W row n0+l15.
    const float* aPtr = A + (m0 + l15) * DIMSZ + half * 2;
    const float* bPtr = W + (n0 + l15) * DIMSZ + half * 2;

    v8f acc = {};
    #pragma unroll 8
    for (int k = 0; k < DIMSZ; k += 4) {
        v2f a = *(const v2f*)(aPtr + k);
        v2f b = *(const v2f*)(bPtr + k);
        // (neg_a, A, neg_b, B, c_mod, C, reuse_a, reuse_b)
        acc = __builtin_amdgcn_wmma_f32_16x16x4_f32(
            false, a, false, b, (short)0, acc, false, false);
    }

    // D layout: lane holds column n0+l15; VGPR r holds row m0 + half*8 + r.
    const float bb   = bias[n0 + l15];
    const int   col  = n0 + l15;
    const int   row0 = m0 + half * 8;
    #pragma unroll
    for (int r = 0; r < 8; ++r) {
        out[(row0 + r) * DIMSZ + col] = acc[r] + bb;
    }
}

// ---------------------------------------------------------------------------
// Broadcast: out[b, t*TPF + i, :] = Y[b*16 + t, :]   (softmax over kv_len==1
// is exactly 1, so the entire q/score path is dead code).
// 204 MB of stores -> pure HBM-bandwidth bound. 128-bit nontemporal stores;
// Y (128 KB) stays hot in cache via regular loads.
// ---------------------------------------------------------------------------
__global__ __launch_bounds__(256) void bcast_rows(
    const v4f* __restrict__ Y,    // [32, 256] float4 rows
    v4f* __restrict__ out,        // [B*N, 256] float4 rows
    int total4)
{
    int idx = blockIdx.x * blockDim.x + threadIdx.x;
    const int stride = gridDim.x * blockDim.x;
    for (; idx < total4; idx += stride) {
        const int d4  = idx & 255;       // 256 float4 per 1024-float row
        const int tok = idx >> 8;        // 0 .. B*N-1
        const int b   = tok / NPB;
        const int n   = tok - b * NPB;
        const int t   = n / TPF;         // frame index
        const v4f val = Y[(b * TFRAMES + t) * 256 + d4];
        __builtin_nontemporal_store(val, &out[idx]);
    }
}

extern "C" void kernel_launch(void* const* d_in, const int* in_sizes, int n_in,
                              void* d_out, int out_size, void* d_ws, size_t ws_size,
                              hipStream_t stream) {
    // setup_inputs order: x, c, Wq, bq, Wk, bk, Wv, bv, Wo, bo
    const float* c_in = (const float*)d_in[1];
    const float* Wv   = (const float*)d_in[6];
    const float* bv   = (const float*)d_in[7];
    const float* Wo   = (const float*)d_in[8];
    const float* bo   = (const float*)d_in[9];

    float* V = (float*)d_ws;           // [32, 1024]
    float* Y = V + BT * DIMSZ;         // [32, 1024]

    // V = c @ Wv^T + bv ; Y = V @ Wo^T + bo  (dependent launches on stream)
    gemm32_wmma_f32<<<16, 256, 0, stream>>>(c_in, Wv, bv, V);
    gemm32_wmma_f32<<<16, 256, 0, stream>>>(V, Wo, bo, Y);

    // Broadcast Y rows across frames: out_size = 2*24960*1024 floats.
    const int total4 = out_size / 4;                 // 12,779,520 float4
    const int threads = 256;
    int blocks = (total4 + threads * 4 - 1) / (threads * 4);  // ~4 elems/thread
    bcast_rows<<<blocks, threads, 0, stream>>>((const v4f*)Y, (v4f*)d_out, total4);
}